// RaySamples_34454227648765
// MI455X (gfx1250) — compile-verified
//
#include <hip/hip_runtime.h>
#include <hip/hip_bf16.h>

// ---------------------------------------------------------------------------
// NeRF ray-sample weights on MI455X (gfx1250), wave32.
//
// Bandwidth-bound (128 MB @ 23.3 TB/s ~= 5.5 us). Scan implemented as blocked
// matmul against triangular ones tiles using V_WMMA_F32_16X16X4_F32 — the
// extra FLOPs are free under the memory roof, and it keeps data on the CDNA5
// matrix path. All global traffic is coalesced 128b NT loads / NT stores.
// ---------------------------------------------------------------------------

typedef float v2f __attribute__((ext_vector_type(2)));
typedef float v8f __attribute__((ext_vector_type(8)));
typedef float f4v __attribute__((ext_vector_type(4)));

#define KSAMP 128         // samples per ray
#define PITCH 132         // LDS row pitch in floats (528B = 33*16B: keeps 16B
                          // alignment and makes strided reads bank-conflict-free)
#define WAVES_PER_BLOCK 4
#define RAYS_PER_WAVE 16  // one 16x128 tile per wave

__global__ __launch_bounds__(WAVES_PER_BLOCK * 32)
void RaySamples_34454227648765_kernel(const float* __restrict__ dens,
                                      const float* __restrict__ delt,
                                      float* __restrict__ weights,
                                      float* __restrict__ trans) {
  __shared__ float lds[WAVES_PER_BLOCK * RAYS_PER_WAVE * PITCH];

  const int lane = threadIdx.x & 31;
  const int wave = threadIdx.x >> 5;
  const int rayBase = (blockIdx.x * WAVES_PER_BLOCK + wave) * RAYS_PER_WAVE;
  float* tile = &lds[wave * RAYS_PER_WAVE * PITCH];

  // ---- Phase 1: stream 16 rays x 128 samples of both inputs (contiguous
  // 8KB slabs), compute dd = delta*density, park dd in padded LDS tile. ----
  const size_t gbase = (size_t)rayBase * KSAMP;           // float index
  const f4v* dp = (const f4v*)(dens + gbase);
  const f4v* tp = (const f4v*)(delt + gbase);
#pragma unroll
  for (int i = 0; i < 16; ++i) {                          // 512 float4 / 32 lanes
    int f  = i * 32 + lane;                               // float4 index in tile
    int m  = f >> 5;                                      // ray row (128f = 32xf4)
    int k4 = f & 31;                                      // float4 column
    f4v a  = __builtin_nontemporal_load(dp + f);
    f4v b  = __builtin_nontemporal_load(tp + f);
    f4v dd = a * b;
    *(f4v*)(tile + m * PITCH + k4 * 4) = dd;              // 16B-aligned ds_store_b128
  }
  __syncthreads();

  const int hi = lane >> 4;    // half-wave: 0 = lanes 0-15, 1 = lanes 16-31
  const int lm = lane & 15;

  // carry[v] = running per-ray sum for row (v + 8*hi), replicated across the
  // 16 column lanes — i.e. exactly the WMMA C-matrix layout.
  v8f carry = {0.f, 0.f, 0.f, 0.f, 0.f, 0.f, 0.f, 0.f};

  for (int blk = 0; blk < 8; ++blk) {                     // 8 column blocks of 16
    const int base = blk * 16;
    v8f acc = carry;

#pragma unroll
    for (int c = 0; c < 4; ++c) {
      // A (16x4 f32): lane holds row m=lm, K = 4c + 2*hi + {0,1}  (b64 LDS read)
      const float* ap = tile + lm * PITCH + base + c * 4 + 2 * hi;
      v2f A = *(const v2f*)ap;
      // B (4x16): triangular ones tile, T[k][n] = (4c+k <= n); generated per lane.
      // Assumed layout mirrors A: vgpr v holds row K = v + 2*hi, col n = lm.
      int k0 = c * 4 + 2 * hi;
      v2f B;
      B.x = (k0     <= lm) ? 1.0f : 0.0f;
      B.y = (k0 + 1 <= lm) ? 1.0f : 0.0f;
      acc = __builtin_amdgcn_wmma_f32_16x16x4_f32(
          /*neg_a=*/false, A, /*neg_b=*/false, B,
          /*c_mod=*/(short)0, acc, /*reuse_a=*/false, /*reuse_b=*/false);
    }
    // acc = carry + inclusive scan within block (C layout: vgpr v, row v+8*hi,
    // col = lm). New carry = column 15 of acc, broadcast across each half-wave.
    v8f newcarry;
#pragma unroll
    for (int v = 0; v < 8; ++v) newcarry[v] = __shfl(acc[v], 15, 16);

    // Emit outputs directly from C layout. Per store: 32 lanes cover two
    // aligned 64B runs (rows are 512B aligned) -> full cacheline utilization.
#pragma unroll
    for (int v = 0; v < 8; ++v) {
      int   m   = v + 8 * hi;
      float ddv = tile[m * PITCH + base + lm];            // conflict-free b32 read
      float s   = acc[v];                                 // inclusive optical depth
      float Tin = __expf(-s);                             // exp(-inclusive)
      float T   = __expf(ddv - s);                        // exp(-exclusive) = transmittance
      float w   = T - Tin;                                // alphas * transmittance
      size_t o  = (size_t)(rayBase + m) * KSAMP + base + lm;
      __builtin_nontemporal_store(w, weights + o);
      __builtin_nontemporal_store(T, trans + o);
    }
    carry = newcarry;
  }
}

extern "C" void kernel_launch(void* const* d_in, const int* in_sizes, int n_in,
                              void* d_out, int out_size, void* d_ws, size_t ws_size,
                              hipStream_t stream) {
  const float* dens = (const float*)d_in[0];
  const float* delt = (const float*)d_in[1];
  const int nrays = in_sizes[0] / KSAMP;                  // 65536
  float* weights = (float*)d_out;                         // [N,K,1] flat
  float* trans   = weights + (size_t)nrays * KSAMP;       // [N,K,1] flat

  const int raysPerBlock = WAVES_PER_BLOCK * RAYS_PER_WAVE;  // 64
  dim3 grid((nrays + raysPerBlock - 1) / raysPerBlock);      // 1024
  dim3 block(WAVES_PER_BLOCK * 32);                          // 128 threads, 4 waves
  RaySamples_34454227648765_kernel<<<grid, block, 0, stream>>>(dens, delt,
                                                               weights, trans);
}